// GINNodeEmbedding_29351806501559
// MI455X (gfx1250) — compile-verified
//
#include <hip/hip_runtime.h>

#define N_NODES 200000
#define N_EDGES 500000
#define EMB 128
#define NLAYERS 5
#define BN_EPS 1e-5f

typedef __attribute__((ext_vector_type(2))) float v2f;
typedef __attribute__((ext_vector_type(8))) float v8f;
typedef __attribute__((ext_vector_type(4))) int   v4i;

// ---- async global->LDS copy (CDNA5 GLOBAL_LOAD_ASYNC_TO_LDS_B128, ASYNCcnt)
#if defined(__has_builtin)
#if __has_builtin(__builtin_amdgcn_global_load_async_to_lds_b128)
#define HAS_ASYNC_LDS 1
#endif
#endif
#ifndef HAS_ASYNC_LDS
#define HAS_ASYNC_LDS 0
#endif

__device__ __forceinline__ void async_copy_b128(const float* g, float* l) {
#if HAS_ASYNC_LDS
    __builtin_amdgcn_global_load_async_to_lds_b128(
        (__attribute__((address_space(1))) v4i*)g,
        (__attribute__((address_space(3))) v4i*)l, 0, 0);
#else
    *(float4*)l = *(const float4*)g;
#endif
}

__device__ __forceinline__ void async_drain() {
#if HAS_ASYNC_LDS
    asm volatile("s_wait_asynccnt 0" ::: "memory");
#endif
}

struct AtomTabs { const float* t[9]; };

// ---------------------------------------------------------------- atom encoder
__global__ __launch_bounds__(256)
void k_atom_embed(const int* __restrict__ x, AtomTabs tabs, float* __restrict__ h) {
    int idx = blockIdx.x * 256 + threadIdx.x;      // N * 32 threads, 4 ch each
    int n = idx >> 5;
    int c = (idx & 31) * 4;
    if (n >= N_NODES) return;
    float4 acc = make_float4(0.f, 0.f, 0.f, 0.f);
#pragma unroll
    for (int i = 0; i < 9; ++i) {
        int row = x[n * 9 + i];
        float4 v = *(const float4*)(tabs.t[i] + row * EMB + c);
        acc.x += v.x; acc.y += v.y; acc.z += v.z; acc.w += v.w;
    }
    *(float4*)(h + n * EMB + c) = acc;
}

// ------------------------------------------------------- agg = (1 + eps) * h
__global__ __launch_bounds__(256)
void k_scale_init(const float* __restrict__ h, const float* __restrict__ epsP,
                  float* __restrict__ agg) {
    int idx = blockIdx.x * 256 + threadIdx.x;      // N*EMB/4 threads
    float s = 1.0f + *epsP;
    float4 v = *(const float4*)(h + idx * 4);
    v.x *= s; v.y *= s; v.z *= s; v.w *= s;
    *(float4*)(agg + idx * 4) = v;
}

// ------------------- edge message: relu(h[src] + bond_emb) scatter-add to agg
__global__ __launch_bounds__(256)
void k_edge(const int* __restrict__ ei, const int* __restrict__ ea,
            const float* __restrict__ h,
            const float* __restrict__ t0, const float* __restrict__ t1,
            const float* __restrict__ t2, float* __restrict__ agg) {
    int idx = blockIdx.x * 256 + threadIdx.x;      // E * 32 threads, 4 ch each
    int e = idx >> 5;
    int c = (idx & 31) * 4;
    if (e >= N_EDGES) return;
    int src = ei[e];
    int dst = ei[N_EDGES + e];
    int a0 = ea[e * 3 + 0], a1 = ea[e * 3 + 1], a2 = ea[e * 3 + 2];
    float4 hv = *(const float4*)(h + src * EMB + c);
    float4 b0 = *(const float4*)(t0 + a0 * EMB + c);
    float4 b1 = *(const float4*)(t1 + a1 * EMB + c);
    float4 b2 = *(const float4*)(t2 + a2 * EMB + c);
    float m0 = fmaxf(hv.x + b0.x + b1.x + b2.x, 0.f);
    float m1 = fmaxf(hv.y + b0.y + b1.y + b2.y, 0.f);
    float m2 = fmaxf(hv.z + b0.z + b1.z + b2.z, 0.f);
    float m3 = fmaxf(hv.w + b0.w + b1.w + b2.w, 0.f);
    float* p = agg + dst * EMB + c;
    atomicAdd(p + 0, m0);
    atomicAdd(p + 1, m1);
    atomicAdd(p + 2, m2);
    atomicAdd(p + 3, m3);
}

// ----------------------------------------------------------- WMMA f32 GEMM
// Out[M,128] = (opt BN+ReLU)(A[M,128]) @ W[128,128] + bias
// block = 256 threads (8 waves), tile = 64 rows x FULL 128 cols (A read once)
#define SA_STRIDE 132   // 4m mod 64 distinct across 16 rows; K+2 shift disjoint
#define SW_STRIDE 136   // 2*136 mod 64 = 16 -> lane16-31 rows on disjoint banks
__global__ __launch_bounds__(256)
void k_gemm(const float* __restrict__ A, const float* __restrict__ W,
            const float* __restrict__ bias,
            const float* __restrict__ bnScale, const float* __restrict__ bnShift,
            int applyBnRelu, float* __restrict__ Out) {
    extern __shared__ float smem[];
    float* sA = smem;                        // [64][SA_STRIDE]
    float* sW = smem + 64 * SA_STRIDE;       // [128][SW_STRIDE]

    const int rowBase = blockIdx.x * 64;
    const int tid = threadIdx.x;

    // stage full W (128x128) via async b128 DMA: 4096 float4 chunks
    for (int f = tid; f < 128 * 32; f += 256) {
        int k = f >> 5, j = (f & 31) * 4;
        async_copy_b128(W + k * EMB + j, sW + k * SW_STRIDE + j);
    }
    // stage A tile (64x128)
    if (applyBnRelu) {
        for (int f = tid; f < 64 * 32; f += 256) {
            int r = f >> 5, c = (f & 31) * 4;
            float4 v  = *(const float4*)(A + (rowBase + r) * EMB + c);
            float4 sc = *(const float4*)(bnScale + c);
            float4 sh = *(const float4*)(bnShift + c);
            v.x = fmaxf(v.x * sc.x + sh.x, 0.f);
            v.y = fmaxf(v.y * sc.y + sh.y, 0.f);
            v.z = fmaxf(v.z * sc.z + sh.z, 0.f);
            v.w = fmaxf(v.w * sc.w + sh.w, 0.f);
            *(float4*)(sA + r * SA_STRIDE + c) = v;
        }
    } else {
        for (int f = tid; f < 64 * 32; f += 256) {
            int r = f >> 5, c = (f & 31) * 4;
            async_copy_b128(A + (rowBase + r) * EMB + c, sA + r * SA_STRIDE + c);
        }
    }
    async_drain();
    __syncthreads();

    const int wave = tid >> 5;
    const int lane = tid & 31;
    const int m0  = (wave & 3) * 16;         // 4 waves across M (64 rows)
    const int n0w = (wave >> 2) * 64;        // 2 waves across N, 4 tiles each
    const int mr  = lane & 15;
    const int kh  = (lane >> 4) * 2;         // lanes 16-31 hold K+2,K+3

    // C init: C[M][N] = bias[N] (fused bias add)
    v8f acc[4];
#pragma unroll
    for (int t = 0; t < 4; ++t) {
        float bv = bias[n0w + 16 * t + mr];
        acc[t] = (v8f){bv, bv, bv, bv, bv, bv, bv, bv};
    }

    const float* aRow = sA + (m0 + mr) * SA_STRIDE;
#pragma unroll
    for (int k = 0; k < EMB; k += 4) {
        v2f a;
        a.x = aRow[k + kh];
        a.y = aRow[k + kh + 1];
#pragma unroll
        for (int t = 0; t < 4; ++t) {
            v2f b;
            b.x = sW[(k + kh) * SW_STRIDE + n0w + 16 * t + mr];
            b.y = sW[(k + kh + 1) * SW_STRIDE + n0w + 16 * t + mr];
            acc[t] = __builtin_amdgcn_wmma_f32_16x16x4_f32(
                false, a, false, b, (short)0, acc[t], false, false);
        }
    }

    // store D: VGPR v -> row m0 + v + 8*(lane>=16), col = lane&15
    const int rowTop = rowBase + m0 + (lane >> 4) * 8;
#pragma unroll
    for (int t = 0; t < 4; ++t)
#pragma unroll
        for (int v = 0; v < 8; ++v)
            Out[(rowTop + v) * EMB + n0w + 16 * t + mr] = acc[t][v];
}

// ------------------------------------------------------------- BN statistics
__global__ __launch_bounds__(256)
void k_zero_stats(float* __restrict__ sums, float* __restrict__ sumsq) {
    int t = threadIdx.x;
    if (t < EMB) sums[t] = 0.f;
    else sumsq[t - EMB] = 0.f;
}

__global__ __launch_bounds__(256)
void k_stats(const float* __restrict__ t, float* __restrict__ sums,
             float* __restrict__ sumsq) {
    int c = threadIdx.x & 127;
    int rowStart = blockIdx.x * 2 + (threadIdx.x >> 7);
    int stride = gridDim.x * 2;
    float s = 0.f, s2 = 0.f;
    for (int r = rowStart; r < N_NODES; r += stride) {
        float v = t[r * EMB + c];
        s += v;
        s2 += v * v;
    }
    atomicAdd(&sums[c], s);
    atomicAdd(&sumsq[c], s2);
}

__global__ __launch_bounds__(128)
void k_bn_finalize(const float* __restrict__ sums, const float* __restrict__ sumsq,
                   const float* __restrict__ g, const float* __restrict__ b,
                   float* __restrict__ scale, float* __restrict__ shift) {
    int c = threadIdx.x;
    const float inv = 1.0f / (float)N_NODES;
    float mean = sums[c] * inv;
    float var = sumsq[c] * inv - mean * mean;   // biased var, matches jnp/torch
    float rs = rsqrtf(var + BN_EPS);
    float sc = g[c] * rs;
    scale[c] = sc;
    shift[c] = b[c] - mean * sc;
}

// --------------------------------------------- h = BN(t2) (+ReLU except last)
__global__ __launch_bounds__(256)
void k_bn_out(const float* __restrict__ t, const float* __restrict__ scale,
              const float* __restrict__ shift, int relu, float* __restrict__ out) {
    int idx = blockIdx.x * 256 + threadIdx.x;   // N*EMB/4 threads
    int c = (idx & 31) * 4;
    float4 v = *(const float4*)(t + idx * 4);
    float4 sc = *(const float4*)(scale + c);
    float4 sh = *(const float4*)(shift + c);
    v.x = v.x * sc.x + sh.x;
    v.y = v.y * sc.y + sh.y;
    v.z = v.z * sc.z + sh.z;
    v.w = v.w * sc.w + sh.w;
    if (relu) {
        v.x = fmaxf(v.x, 0.f); v.y = fmaxf(v.y, 0.f);
        v.z = fmaxf(v.z, 0.f); v.w = fmaxf(v.w, 0.f);
    }
    *(float4*)(out + idx * 4) = v;
}

// ----------------------------------------------------------------- launcher
extern "C" void kernel_launch(void* const* d_in, const int* in_sizes, int n_in,
                              void* d_out, int out_size, void* d_ws, size_t ws_size,
                              hipStream_t stream) {
    (void)in_sizes; (void)n_in; (void)out_size; (void)ws_size;

    const int* x  = (const int*)d_in[0];
    const int* ei = (const int*)d_in[1];
    const int* ea = (const int*)d_in[2];
    AtomTabs tabs;
    for (int i = 0; i < 9; ++i) tabs.t[i] = (const float*)d_in[3 + i];

    // workspace layout
    float* h     = (float*)d_ws;
    float* agg   = h + (size_t)N_NODES * EMB;      // also reused as t2
    float* t1    = agg + (size_t)N_NODES * EMB;
    float* sums  = t1 + (size_t)N_NODES * EMB;
    float* sumsq = sums + EMB;
    float* scale = sumsq + EMB;
    float* shift = scale + EMB;
    float* outH  = (float*)d_out;

    const dim3 blk(256);
    const dim3 gNode((N_NODES * 32) / 256);        // 25000
    const dim3 gElem((N_NODES * EMB / 4) / 256);   // 25000
    const dim3 gEdge((N_EDGES * 32) / 256);        // 62500
    const dim3 gGemm(N_NODES / 64);                // 3125, full 128-col tile
    const size_t shm = (64 * SA_STRIDE + 128 * SW_STRIDE) * sizeof(float); // 103424 B

    k_atom_embed<<<gNode, blk, 0, stream>>>(x, tabs, h);

    for (int l = 0; l < NLAYERS; ++l) {
        const int base = 12 + l * 12;
        const float* bt0  = (const float*)d_in[base + 0];
        const float* bt1  = (const float*)d_in[base + 1];
        const float* bt2  = (const float*)d_in[base + 2];
        const float* W1   = (const float*)d_in[base + 3];
        const float* b1   = (const float*)d_in[base + 4];
        const float* g1   = (const float*)d_in[base + 5];
        const float* be1  = (const float*)d_in[base + 6];
        const float* W2   = (const float*)d_in[base + 7];
        const float* b2   = (const float*)d_in[base + 8];
        const float* epsP = (const float*)d_in[base + 9];
        const float* g2   = (const float*)d_in[base + 10];
        const float* be2  = (const float*)d_in[base + 11];

        k_scale_init<<<gElem, blk, 0, stream>>>(h, epsP, agg);
        k_edge<<<gEdge, blk, 0, stream>>>(ei, ea, h, bt0, bt1, bt2, agg);

        // t1 = agg @ W1 + b1
        k_gemm<<<gGemm, blk, shm, stream>>>(agg, W1, b1, nullptr, nullptr, 0, t1);
        k_zero_stats<<<1, 256, 0, stream>>>(sums, sumsq);
        k_stats<<<512, blk, 0, stream>>>(t1, sums, sumsq);
        k_bn_finalize<<<1, 128, 0, stream>>>(sums, sumsq, g1, be1, scale, shift);

        // t2 (alias agg) = relu(bn1(t1)) @ W2 + b2
        k_gemm<<<gGemm, blk, shm, stream>>>(t1, W2, b2, scale, shift, 1, agg);
        k_zero_stats<<<1, 256, 0, stream>>>(sums, sumsq);
        k_stats<<<512, blk, 0, stream>>>(agg, sums, sumsq);
        k_bn_finalize<<<1, 128, 0, stream>>>(sums, sumsq, g2, be2, scale, shift);

        float* hdst = (l == NLAYERS - 1) ? outH : h;
        k_bn_out<<<gElem, blk, 0, stream>>>(agg, scale, shift,
                                            (l < NLAYERS - 1) ? 1 : 0, hdst);
    }
}